// Net_5600637354347
// MI455X (gfx1250) — compile-verified
//
#include <hip/hip_runtime.h>
#include <math.h>

typedef __attribute__((ext_vector_type(2))) float v2f;
typedef __attribute__((ext_vector_type(8))) float v8f;

#define NEG_SLOPE 0.2f

// ---------- helpers ----------
__device__ __forceinline__ unsigned fkey_enc(float f) {
  unsigned u = __float_as_uint(f);
  return (u & 0x80000000u) ? ~u : (u | 0x80000000u);   // monotonic order-preserving key
}
__device__ __forceinline__ float fkey_dec(unsigned u) {
  return (u & 0x80000000u) ? __uint_as_float(u & 0x7FFFFFFFu) : __uint_as_float(~u);
}
// edge id -> (src,dst) with implicit self-loops appended after the E real edges
__device__ __forceinline__ void edge_sd(const int* __restrict__ ei, int E, int e, int& s, int& d) {
  if (e < E) { s = ei[e]; d = ei[E + e]; } else { s = e - E; d = e - E; }
}

// ---------- fp32 WMMA GEMM: C[M,Nc] = A[M,K] * B[K,Nc] ----------
// One wave computes a 16 x (16*TN) strip: the A fragment is loaded once per
// k-step and fed to TN back-to-back v_wmma_f32_16x16x4_f32 ops (register
// blocking along N). K and Nc are compile-time so all load offsets fold into
// immediate byte offsets (kills per-iteration 64-bit address VALU).
template <int TN, int K, int NC>
__global__ void gemm_wmma_f32(const float* __restrict__ A, const float* __restrict__ B,
                              float* __restrict__ C, int M) {
  constexpr int GROUPS = NC / (16 * TN);               // column strips
  int wid  = (int)((blockIdx.x * blockDim.x + threadIdx.x) >> 5);
  int lane = (int)(threadIdx.x & 31);
  int nstrips = (M >> 4) * GROUPS;
  if (wid >= nstrips) return;                          // wave-uniform: EXEC stays all-ones
  int tm = wid / GROUPS, tg = wid % GROUPS;
  int half = lane >> 4;                                // half 0 -> K+0/1, half 1 -> K+2/3
  int l    = lane & 15;

  const float* arow = A + (size_t)(tm * 16 + l) * K + 2 * half;
  const float* bcol = B + (size_t)(2 * half) * NC + tg * (16 * TN) + l;

  v8f c[TN];
#pragma unroll
  for (int t = 0; t < TN; ++t) c[t] = (v8f){};

#pragma unroll
  for (int k = 0; k < K; k += 4) {
    v2f a;
    a[0] = arow[k + 0];
    a[1] = arow[k + 1];
#pragma unroll
    for (int t = 0; t < TN; ++t) {
      v2f b;
      b[0] = bcol[(size_t)(k + 0) * NC + t * 16];
      b[1] = bcol[(size_t)(k + 1) * NC + t * 16];
      // (neg_a, A, neg_b, B, c_mod, C, reuse_a, reuse_b)
      c[t] = __builtin_amdgcn_wmma_f32_16x16x4_f32(false, a, false, b, (short)0, c[t],
                                                   false, false);
    }
  }

  float* crow = C + (size_t)(tm * 16 + half * 8) * NC + tg * (16 * TN) + l;
#pragma unroll
  for (int t = 0; t < TN; ++t)
#pragma unroll
    for (int r = 0; r < 8; ++r) crow[(size_t)r * NC + t * 16] = c[t][r];
}

// ---------- layer-1 attention logits: al_s/al_d[n,h] = <h1[n,h,:], a[h,:]> ----------
__global__ void attn_logits1(const float* __restrict__ h1, const float* __restrict__ a_src,
                             const float* __restrict__ a_dst, float* __restrict__ al_s,
                             float* __restrict__ al_d, int N) {
  int gid = blockIdx.x * blockDim.x + threadIdx.x;   // node*4 + head
  if (gid >= N * 4) return;
  int n = gid >> 2, h = gid & 3;
  const float* hv = h1 + (size_t)n * 128 + h * 32;
  const float* as = a_src + h * 32;
  const float* ad = a_dst + h * 32;
  float ss = 0.f, sd = 0.f;
#pragma unroll
  for (int i = 0; i < 32; ++i) { float v = hv[i]; ss += v * as[i]; sd += v * ad[i]; }
  al_s[gid] = ss;
  al_d[gid] = sd;
}

// ---------- layer-1 edge pass 1: e = leaky_relu(al_s[s]+al_d[d]); segment max ----------
__global__ void edge_logits1(const int* __restrict__ ei, int E, int Etot,
                             const float* __restrict__ al_s, const float* __restrict__ al_d,
                             float* __restrict__ ebuf, unsigned* __restrict__ mkey) {
  int gid = blockIdx.x * blockDim.x + threadIdx.x;   // edge*4 + head
  if (gid >= Etot * 4) return;
  int e = gid >> 2, h = gid & 3;
  int s, d; edge_sd(ei, E, e, s, d);
  float v = al_s[s * 4 + h] + al_d[d * 4 + h];
  v = v > 0.f ? v : NEG_SLOPE * v;
  ebuf[gid] = v;
  atomicMax(&mkey[d * 4 + h], fkey_enc(v));
}

// ---------- layer-1 edge pass 2: ex = exp(e - m[dst]); segment sum ----------
__global__ void edge_exp1(const int* __restrict__ ei, int E, int Etot,
                          const unsigned* __restrict__ mkey,
                          float* __restrict__ ebuf, float* __restrict__ den) {
  int gid = blockIdx.x * blockDim.x + threadIdx.x;
  if (gid >= Etot * 4) return;
  int e = gid >> 2, h = gid & 3;
  int s, d; edge_sd(ei, E, e, s, d);
  float ex = expf(ebuf[gid] - fkey_dec(mkey[d * 4 + h]));
  ebuf[gid] = ex;
  atomicAdd(&den[d * 4 + h], ex);
}

// ---------- layer-1 edge pass 3: acc[dst] += h1[src] * alpha  (wave per edge, 4 feats/lane) ----
__global__ void edge_scatter1(const int* __restrict__ ei, int E, int Etot,
                              const float* __restrict__ h1, const float* __restrict__ ex,
                              const float* __restrict__ den, float* __restrict__ acc) {
  int wid  = (int)((blockIdx.x * blockDim.x + threadIdx.x) >> 5);
  int lane = (int)(threadIdx.x & 31);
  if (wid >= Etot) return;
  int s, d; edge_sd(ei, E, wid, s, d);
  int h = lane >> 3;                                  // feature base lane*4 -> head = lane/8
  float alpha = ex[wid * 4 + h] / den[d * 4 + h];
  const float4 hv = *(const float4*)(h1 + (size_t)s * 128 + lane * 4);
  float* o = acc + (size_t)d * 128 + lane * 4;
  atomicAdd(o + 0, hv.x * alpha);
  atomicAdd(o + 1, hv.y * alpha);
  atomicAdd(o + 2, hv.z * alpha);
  atomicAdd(o + 3, hv.w * alpha);
}

// ---------- layer-1 finalize: acc = elu(acc + b1) ----------
__global__ void finalize1(float* __restrict__ acc, const float* __restrict__ b1, int N) {
  int gid = blockIdx.x * blockDim.x + threadIdx.x;
  if (gid >= N * 128) return;
  float v = acc[gid] + b1[gid & 127];
  acc[gid] = v > 0.f ? v : (expf(v) - 1.f);
}

// ---------- layer-2 attention logits (1 head, 16 dims) ----------
__global__ void attn_logits2(const float* __restrict__ h2, const float* __restrict__ a_src,
                             const float* __restrict__ a_dst, float* __restrict__ al_s,
                             float* __restrict__ al_d, int N) {
  int n = blockIdx.x * blockDim.x + threadIdx.x;
  if (n >= N) return;
  float ss = 0.f, sd = 0.f;
#pragma unroll
  for (int c = 0; c < 16; ++c) { float v = h2[(size_t)n * 16 + c]; ss += v * a_src[c]; sd += v * a_dst[c]; }
  al_s[n] = ss;
  al_d[n] = sd;
}

__global__ void edge_logits2(const int* __restrict__ ei, int E, int Etot,
                             const float* __restrict__ al_s, const float* __restrict__ al_d,
                             float* __restrict__ ebuf, unsigned* __restrict__ mkey) {
  int e = blockIdx.x * blockDim.x + threadIdx.x;
  if (e >= Etot) return;
  int s, d; edge_sd(ei, E, e, s, d);
  float v = al_s[s] + al_d[d];
  v = v > 0.f ? v : NEG_SLOPE * v;
  ebuf[e] = v;
  atomicMax(&mkey[d], fkey_enc(v));
}

__global__ void edge_exp2(const int* __restrict__ ei, int E, int Etot,
                          const unsigned* __restrict__ mkey,
                          float* __restrict__ ebuf, float* __restrict__ den) {
  int e = blockIdx.x * blockDim.x + threadIdx.x;
  if (e >= Etot) return;
  int s, d; edge_sd(ei, E, e, s, d);
  float ex = expf(ebuf[e] - fkey_dec(mkey[d]));
  ebuf[e] = ex;
  atomicAdd(&den[d], ex);
}

__global__ void edge_scatter2(const int* __restrict__ ei, int E, int Etot,
                              const float* __restrict__ h2, const float* __restrict__ ex,
                              const float* __restrict__ den, float* __restrict__ out) {
  int gid = blockIdx.x * blockDim.x + threadIdx.x;    // edge*16 + class
  if (gid >= Etot * 16) return;
  int e = gid >> 4, f = gid & 15;
  int s, d; edge_sd(ei, E, e, s, d);
  float alpha = ex[e] / den[d];
  atomicAdd(&out[(size_t)d * 16 + f], h2[(size_t)s * 16 + f] * alpha);
}

// ---------- layer-2 finalize: out = log_softmax(out + b2) ----------
__global__ void finalize2(float* __restrict__ out, const float* __restrict__ b2, int N) {
  int n = blockIdx.x * blockDim.x + threadIdx.x;
  if (n >= N) return;
  float v[16];
  float mx = -INFINITY;
#pragma unroll
  for (int c = 0; c < 16; ++c) { v[c] = out[(size_t)n * 16 + c] + b2[c]; mx = fmaxf(mx, v[c]); }
  float sum = 0.f;
#pragma unroll
  for (int c = 0; c < 16; ++c) sum += expf(v[c] - mx);
  float lse = mx + logf(sum);
#pragma unroll
  for (int c = 0; c < 16; ++c) out[(size_t)n * 16 + c] = v[c] - lse;
}

// ---------- launch ----------
extern "C" void kernel_launch(void* const* d_in, const int* in_sizes, int n_in,
                              void* d_out, int out_size, void* d_ws, size_t ws_size,
                              hipStream_t stream) {
  const float* x      = (const float*)d_in[0];
  const int*   ei     = (const int*)d_in[1];
  const float* W1     = (const float*)d_in[2];
  const float* a_src1 = (const float*)d_in[3];
  const float* a_dst1 = (const float*)d_in[4];
  const float* b1     = (const float*)d_in[5];
  const float* W2     = (const float*)d_in[6];
  const float* a_src2 = (const float*)d_in[7];
  const float* a_dst2 = (const float*)d_in[8];
  const float* b2     = (const float*)d_in[9];
  float* out = (float*)d_out;

  const int N    = in_sizes[0] / 128;   // 50000
  const int E    = in_sizes[1] / 2;     // 800000
  const int Etot = E + N;               // +self loops

  // carve workspace
  char* p = (char*)d_ws;
  auto carve = [&](size_t bytes) -> void* {
    void* r = (void*)p;
    p += (bytes + 255) & ~(size_t)255;
    return r;
  };
  float*    h1   = (float*)carve((size_t)N * 128 * 4);
  float*    als1 = (float*)carve((size_t)N * 4 * 4);
  float*    ald1 = (float*)carve((size_t)N * 4 * 4);
  unsigned* m1   = (unsigned*)carve((size_t)N * 4 * 4);
  float*    den1 = (float*)carve((size_t)N * 4 * 4);
  float*    ex1  = (float*)carve((size_t)Etot * 4 * 4);
  float*    acc1 = (float*)carve((size_t)N * 128 * 4);
  float*    h2   = (float*)carve((size_t)N * 16 * 4);
  float*    als2 = (float*)carve((size_t)N * 4);
  float*    ald2 = (float*)carve((size_t)N * 4);
  unsigned* m2   = (unsigned*)carve((size_t)N * 4);
  float*    den2 = (float*)carve((size_t)N * 4);
  float*    ex2  = (float*)carve((size_t)Etot * 4);

  hipMemsetAsync(m1,   0, (size_t)N * 4 * 4,   stream);
  hipMemsetAsync(den1, 0, (size_t)N * 4 * 4,   stream);
  hipMemsetAsync(acc1, 0, (size_t)N * 128 * 4, stream);
  hipMemsetAsync(m2,   0, (size_t)N * 4,       stream);
  hipMemsetAsync(den2, 0, (size_t)N * 4,       stream);
  hipMemsetAsync(out,  0, (size_t)N * 16 * 4,  stream);

  const int T = 256;
  // ---- layer 1: h1 = x @ W1 (N x 128 x 128), 16x64 strip per wave ----
  {
    int waves = (N / 16) * (128 / 64);               // TN=4
    gemm_wmma_f32<4, 128, 128><<<(waves + 7) / 8, T, 0, stream>>>(x, W1, h1, N);
  }
  attn_logits1<<<(N * 4 + T - 1) / T, T, 0, stream>>>(h1, a_src1, a_dst1, als1, ald1, N);
  edge_logits1<<<(Etot * 4 + T - 1) / T, T, 0, stream>>>(ei, E, Etot, als1, ald1, ex1, m1);
  edge_exp1<<<(Etot * 4 + T - 1) / T, T, 0, stream>>>(ei, E, Etot, m1, ex1, den1);
  edge_scatter1<<<(Etot + 7) / 8, T, 0, stream>>>(ei, E, Etot, h1, ex1, den1, acc1);
  finalize1<<<(N * 128 + T - 1) / T, T, 0, stream>>>(acc1, b1, N);

  // ---- layer 2: h2 = acc1 @ W2 (N x 128 x 16), 16x16 tile per wave ----
  {
    int waves = (N / 16);                            // TN=1
    gemm_wmma_f32<1, 128, 16><<<(waves + 7) / 8, T, 0, stream>>>(acc1, W2, h2, N);
  }
  attn_logits2<<<(N + T - 1) / T, T, 0, stream>>>(h2, a_src2, a_dst2, als2, ald2, N);
  edge_logits2<<<(Etot + T - 1) / T, T, 0, stream>>>(ei, E, Etot, als2, ald2, ex2, m2);
  edge_exp2<<<(Etot + T - 1) / T, T, 0, stream>>>(ei, E, Etot, m2, ex2, den2);
  edge_scatter2<<<(Etot * 16 + T - 1) / T, T, 0, stream>>>(ei, E, Etot, h2, ex2, den2, out);
  finalize2<<<(N + T - 1) / T, T, 0, stream>>>(out, b2, N);
}